// Decoder_41618233099072
// MI455X (gfx1250) — compile-verified
//
#include <hip/hip_runtime.h>
#include <math.h>

typedef __attribute__((ext_vector_type(2))) float v2f;
typedef __attribute__((ext_vector_type(8))) float v8f;

#define B_DIM 64
#define D_DIM 256
#define H_DIM 512
#define G_DIM 2048   // 4*H
#define V_DIM 32000
#define SEQ   64

// ---------------------------------------------------------------------------
// WMMA fp32 operand layouts (ISA 7.12.2), wave32:
//   A 16x4 tile : lane = hi*16+lr holds {A[m0+lr][k0+2*hi], A[m0+lr][k0+2*hi+1]}
//   B 4x16 tile : lane = hi*16+lr holds {B[k0+2*hi][n0+lr], B[k0+2*hi+1][n0+lr]}
// We pre-pack both operands so each lane reads ONE contiguous float2:
//   Bpack[((ntile*KB + kblk)*32 + lane)*2]   (KB = K/4)   -- per-ntile K-walk
//   Apack[((kblk*4  + mtile)*32 + lane)*2]   (4 mtiles over B=64)
// ---------------------------------------------------------------------------

// Pack weight W[N][K] (row-major, N = output dim) into Bpack.
__global__ __launch_bounds__(256)
void pack_b_kernel(const float* __restrict__ W, float* __restrict__ dst,
                   int N, int K) {
    const size_t t = (size_t)blockIdx.x * blockDim.x + threadIdx.x; // N*K/2 threads
    const int KB   = K >> 2;
    const int lane = (int)(t & 31);
    const int kblk = (int)((t >> 5) % KB);
    const int ntile= (int)(t / ((size_t)32 * KB));
    const int lr   = lane & 15;
    const int koff = (lane >> 4) << 1;
    v2f v = *(const v2f*)(W + (size_t)(ntile * 16 + lr) * K + kblk * 4 + koff);
    *(v2f*)(dst + t * 2) = v;
}

// Pack activation src[64][K] into Apack (used once for h0).
__global__ __launch_bounds__(256)
void pack_a_kernel(const float* __restrict__ src, float* __restrict__ dst,
                   int K) {
    const size_t t = (size_t)blockIdx.x * blockDim.x + threadIdx.x; // 64*K/2 threads
    const int lane  = (int)(t & 31);
    const int mtile = (int)((t >> 5) & 3);
    const int kblk  = (int)(t >> 7);
    const int lr    = lane & 15;
    const int koff  = (lane >> 4) << 1;
    v2f v = *(const v2f*)(src + (size_t)(mtile * 16 + lr) * K + kblk * 4 + koff);
    *(v2f*)(dst + t * 2) = v;
}

__global__ void zero_kernel(float* __restrict__ p, int n) {
    int i = blockIdx.x * blockDim.x + threadIdx.x;
    if (i < n) p[i] = 0.f;
}

// ---------------------------------------------------------------------------
// Gates GEMM (fp32 WMMA): gates[64][2048] = x@Wih^T + h@Whh^T + b_ih + b_hh
// One wave owns one 16-wide N-tile and all of M=64 (4 accumulators).
// 128 waves = 16 blocks x 8 waves. Per K-step: 1 B load + 4 A loads + 4 WMMA.
// ---------------------------------------------------------------------------
__global__ __launch_bounds__(256)
void gates_kernel(const float* __restrict__ Xpack,   // A-pack of x   [B,D]
                  const float* __restrict__ Hpack,   // A-pack of h   [B,H]
                  const float* __restrict__ Bih,     // B-pack of Wih [G,D]
                  const float* __restrict__ Bhh,     // B-pack of Whh [G,H]
                  const float* __restrict__ b_ih,
                  const float* __restrict__ b_hh,
                  float* __restrict__ gates)         // [B, G]
{
    const int lane  = threadIdx.x & 31;
    const int wave  = (blockIdx.x * blockDim.x + threadIdx.x) >> 5;
    const int ntile = wave;                 // 0..127
    const int lr    = lane & 15;

    v8f acc0 = {}, acc1 = {}, acc2 = {}, acc3 = {};

    // ---- phase 1: x @ Wih^T, K = 256 (64 k-blocks) ----
    {
        const float* bp = Bih + ((size_t)ntile * 64) * 64 + lane * 2;
        const float* ap = Xpack + lane * 2;
        #pragma unroll 4
        for (int kb = 0; kb < 64; ++kb) {
            v2f b  = *(const v2f*)(bp + (size_t)kb * 64);
            v2f a0 = *(const v2f*)(ap + (size_t)(kb * 4 + 0) * 64);
            v2f a1 = *(const v2f*)(ap + (size_t)(kb * 4 + 1) * 64);
            v2f a2 = *(const v2f*)(ap + (size_t)(kb * 4 + 2) * 64);
            v2f a3 = *(const v2f*)(ap + (size_t)(kb * 4 + 3) * 64);
            acc0 = __builtin_amdgcn_wmma_f32_16x16x4_f32(false, a0, false, b, (short)0, acc0, false, false);
            acc1 = __builtin_amdgcn_wmma_f32_16x16x4_f32(false, a1, false, b, (short)0, acc1, false, false);
            acc2 = __builtin_amdgcn_wmma_f32_16x16x4_f32(false, a2, false, b, (short)0, acc2, false, false);
            acc3 = __builtin_amdgcn_wmma_f32_16x16x4_f32(false, a3, false, b, (short)0, acc3, false, false);
        }
    }
    // ---- phase 2: h @ Whh^T, K = 512 (128 k-blocks) ----
    {
        const float* bp = Bhh + ((size_t)ntile * 128) * 64 + lane * 2;
        const float* ap = Hpack + lane * 2;
        #pragma unroll 4
        for (int kb = 0; kb < 128; ++kb) {
            v2f b  = *(const v2f*)(bp + (size_t)kb * 64);
            v2f a0 = *(const v2f*)(ap + (size_t)(kb * 4 + 0) * 64);
            v2f a1 = *(const v2f*)(ap + (size_t)(kb * 4 + 1) * 64);
            v2f a2 = *(const v2f*)(ap + (size_t)(kb * 4 + 2) * 64);
            v2f a3 = *(const v2f*)(ap + (size_t)(kb * 4 + 3) * 64);
            acc0 = __builtin_amdgcn_wmma_f32_16x16x4_f32(false, a0, false, b, (short)0, acc0, false, false);
            acc1 = __builtin_amdgcn_wmma_f32_16x16x4_f32(false, a1, false, b, (short)0, acc1, false, false);
            acc2 = __builtin_amdgcn_wmma_f32_16x16x4_f32(false, a2, false, b, (short)0, acc2, false, false);
            acc3 = __builtin_amdgcn_wmma_f32_16x16x4_f32(false, a3, false, b, (short)0, acc3, false, false);
        }
    }
    const int n = ntile * 16 + lr;
    const float bias = b_ih[n] + b_hh[n];
    const int rbase = (lane >> 4) << 3;          // +8 rows for lanes 16-31
    v8f accs[4] = {acc0, acc1, acc2, acc3};
    #pragma unroll
    for (int m = 0; m < 4; ++m) {
        #pragma unroll
        for (int r = 0; r < 8; ++r)
            gates[(size_t)(m * 16 + rbase + r) * G_DIM + n] = accs[m][r] + bias;
    }
}

// ---------------------------------------------------------------------------
// LSTM pointwise. Writes h directly in packed-A layout (no repack kernel).
// ---------------------------------------------------------------------------
__global__ __launch_bounds__(256)
void lstm_kernel(const float* __restrict__ gates,  // [B, 4H]
                 float* __restrict__ c,            // [B, H] linear, in/out
                 float* __restrict__ Hpack)        // A-pack of h [B, H]
{
    const int idx = blockIdx.x * blockDim.x + threadIdx.x;  // B*H threads
    const int b = idx / H_DIM;
    const int j = idx - b * H_DIM;
    const float* g = gates + (size_t)b * G_DIM;
    const float si = 1.f / (1.f + expf(-g[j]));
    const float sf = 1.f / (1.f + expf(-g[H_DIM + j]));
    const float so = 1.f / (1.f + expf(-g[3 * H_DIM + j]));
    const float tg = tanhf(g[2 * H_DIM + j]);
    const float c1 = sf * c[idx] + si * tg;
    c[idx] = c1;
    const float h1 = so * tanhf(c1);
    // packed-A position of (b, j)
    const int kblk  = j >> 2;
    const int lane  = (((j >> 1) & 1) << 4) + (b & 15);
    const int mtile = b >> 4;
    Hpack[((size_t)(kblk * 4 + mtile) * 32 + lane) * 2 + (j & 1)] = h1;
}

// ---------------------------------------------------------------------------
// Logits GEMM (fp32 WMMA): out[b][step][v] = h@Wout^T + b_out
// One wave per 16-wide N-tile, all of M=64. 2000 waves = 250 blocks.
// ---------------------------------------------------------------------------
__global__ __launch_bounds__(256)
void logits_kernel(const float* __restrict__ Hpack,  // A-pack of h [B, H]
                   const float* __restrict__ Bout,   // B-pack of Wout [V, H]
                   const float* __restrict__ b_out,  // [V]
                   float* __restrict__ out,          // [B, S, V]
                   int step)
{
    const int lane  = threadIdx.x & 31;
    const int wave  = (blockIdx.x * blockDim.x + threadIdx.x) >> 5;
    const int ntile = wave;                 // 0..1999
    const int lr    = lane & 15;

    v8f acc0 = {}, acc1 = {}, acc2 = {}, acc3 = {};
    const float* bp = Bout + ((size_t)ntile * 128) * 64 + lane * 2;
    const float* ap = Hpack + lane * 2;
    #pragma unroll 4
    for (int kb = 0; kb < 128; ++kb) {
        v2f b  = *(const v2f*)(bp + (size_t)kb * 64);
        v2f a0 = *(const v2f*)(ap + (size_t)(kb * 4 + 0) * 64);
        v2f a1 = *(const v2f*)(ap + (size_t)(kb * 4 + 1) * 64);
        v2f a2 = *(const v2f*)(ap + (size_t)(kb * 4 + 2) * 64);
        v2f a3 = *(const v2f*)(ap + (size_t)(kb * 4 + 3) * 64);
        acc0 = __builtin_amdgcn_wmma_f32_16x16x4_f32(false, a0, false, b, (short)0, acc0, false, false);
        acc1 = __builtin_amdgcn_wmma_f32_16x16x4_f32(false, a1, false, b, (short)0, acc1, false, false);
        acc2 = __builtin_amdgcn_wmma_f32_16x16x4_f32(false, a2, false, b, (short)0, acc2, false, false);
        acc3 = __builtin_amdgcn_wmma_f32_16x16x4_f32(false, a3, false, b, (short)0, acc3, false, false);
    }
    const int n = ntile * 16 + lr;
    const float bias = b_out[n];
    const int rbase = (lane >> 4) << 3;
    v8f accs[4] = {acc0, acc1, acc2, acc3};
    #pragma unroll
    for (int m = 0; m < 4; ++m) {
        #pragma unroll
        for (int r = 0; r < 8; ++r) {
            const int brow = m * 16 + rbase + r;   // batch index
            out[((size_t)brow * SEQ + step) * V_DIM + n] = accs[m][r] + bias;
        }
    }
}

// ---------------------------------------------------------------------------
// Per-row argmax over V (first-occurrence ties) + embedding gather.
// Writes x directly in packed-A layout for the next gates GEMM.
// ---------------------------------------------------------------------------
__global__ __launch_bounds__(256)
void argmax_embed_kernel(const float* __restrict__ out,    // [B, S, V]
                         int step,
                         const float* __restrict__ embed,  // [V, D]
                         float* __restrict__ Xpack)        // A-pack of x [B, D]
{
    __shared__ float sval[256];
    __shared__ int   sidx[256];
    const int b   = blockIdx.x;
    const int tid = threadIdx.x;
    const float* row = out + ((size_t)b * SEQ + step) * V_DIM;

    float best = -INFINITY;
    int   bidx = 0;
    for (int v = tid; v < V_DIM; v += 256) {
        const float val = row[v];
        if (val > best) { best = val; bidx = v; }   // strictly > keeps lowest idx
    }
    sval[tid] = best; sidx[tid] = bidx;
    __syncthreads();
    for (int off = 128; off > 0; off >>= 1) {
        if (tid < off) {
            const float v2 = sval[tid + off];
            const int   i2 = sidx[tid + off];
            if (v2 > sval[tid] || (v2 == sval[tid] && i2 < sidx[tid])) {
                sval[tid] = v2; sidx[tid] = i2;
            }
        }
        __syncthreads();
    }
    const int widx = sidx[0];
    // D_DIM == 256 == blockDim: one element per thread
    const int d = tid;
    const float e = embed[(size_t)widx * D_DIM + d];
    const int kblk  = d >> 2;
    const int lane  = (((d >> 1) & 1) << 4) + (b & 15);
    const int mtile = b >> 4;
    Xpack[((size_t)(kblk * 4 + mtile) * 32 + lane) * 2 + (d & 1)] = e;
}

// ---------------------------------------------------------------------------
// Host orchestration (graph-capture safe).
// Inputs: 0=batch_size, 1=h0, 2=c0, 3=embed_table, 4=W_ih, 5=W_hh,
//         6=b_ih, 7=b_hh, 8=W_out, 9=b_out
// ---------------------------------------------------------------------------
extern "C" void kernel_launch(void* const* d_in, const int* in_sizes, int n_in,
                              void* d_out, int out_size, void* d_ws, size_t ws_size,
                              hipStream_t stream) {
    (void)in_sizes; (void)n_in; (void)out_size; (void)ws_size;
    const float* h0    = (const float*)d_in[1];
    const float* c0    = (const float*)d_in[2];
    const float* embed = (const float*)d_in[3];
    const float* W_ih  = (const float*)d_in[4];   // [4H, D]
    const float* W_hh  = (const float*)d_in[5];   // [4H, H]
    const float* b_ih  = (const float*)d_in[6];
    const float* b_hh  = (const float*)d_in[7];
    const float* W_out = (const float*)d_in[8];   // [V, H]
    const float* b_out = (const float*)d_in[9];
    float* out = (float*)d_out;

    // Workspace (fp32), ~72.7 MB
    float* ws    = (float*)d_ws;
    float* Bih   = ws;  ws += (size_t)G_DIM * D_DIM;   // packed Wih
    float* Bhh   = ws;  ws += (size_t)G_DIM * H_DIM;   // packed Whh
    float* Bout  = ws;  ws += (size_t)V_DIM * H_DIM;   // packed Wout
    float* Xpack = ws;  ws += (size_t)B_DIM * D_DIM;   // packed x
    float* Hpack = ws;  ws += (size_t)B_DIM * H_DIM;   // packed h
    float* cbuf  = ws;  ws += (size_t)B_DIM * H_DIM;   // linear c
    float* gates = ws;  ws += (size_t)B_DIM * G_DIM;   // linear gates

    // One-time weight packs into WMMA lane layout.
    pack_b_kernel<<<(G_DIM * D_DIM / 2) / 256, 256, 0, stream>>>(W_ih, Bih, G_DIM, D_DIM);
    pack_b_kernel<<<(G_DIM * H_DIM / 2) / 256, 256, 0, stream>>>(W_hh, Bhh, G_DIM, H_DIM);
    pack_b_kernel<<<((size_t)V_DIM * H_DIM / 2) / 256, 256, 0, stream>>>(W_out, Bout, V_DIM, H_DIM);

    // Initial state: x = 0 (packed), h = pack(h0), c = c0
    zero_kernel<<<(B_DIM * D_DIM + 255) / 256, 256, 0, stream>>>(Xpack, B_DIM * D_DIM);
    pack_a_kernel<<<(B_DIM * H_DIM / 2) / 256, 256, 0, stream>>>(h0, Hpack, H_DIM);
    hipMemcpyAsync(cbuf, c0, sizeof(float) * B_DIM * H_DIM, hipMemcpyDeviceToDevice, stream);

    for (int s = 0; s < SEQ; ++s) {
        // 128 waves exactly (16 blocks x 8 waves), EXEC all-1s for WMMA.
        gates_kernel<<<16, 256, 0, stream>>>(Xpack, Hpack, Bih, Bhh, b_ih, b_hh, gates);
        lstm_kernel<<<(B_DIM * H_DIM) / 256, 256, 0, stream>>>(gates, cbuf, Hpack);
        // 2000 waves exactly (250 blocks x 8 waves).
        logits_kernel<<<250, 256, 0, stream>>>(Hpack, Bout, b_out, out, s);
        if (s + 1 < SEQ)
            argmax_embed_kernel<<<B_DIM, 256, 0, stream>>>(out, s, embed, Xpack);
    }
}